// NT_Xent_46471546143019
// MI455X (gfx1250) — compile-verified
//
#include <hip/hip_runtime.h>
#include <math.h>

// ---------------------------------------------------------------------------
// NT-Xent (SimCLR) fused on gfx1250:
//   K1: normalize rows of concat(z1,z2), store in pair-parity-shuffled layout
//   K2: fused zn*zn^T GEMM via V_WMMA_F32_16X16X4_F32 + per-row sum(exp) / pos
//   K3: deterministic final reduction -> mean(log(sum_exp) - pos)
// ---------------------------------------------------------------------------

typedef __attribute__((ext_vector_type(2))) float v2f;
typedef __attribute__((ext_vector_type(4))) float v4f;
typedef __attribute__((ext_vector_type(8))) float v8f;

#define NSAMP      4096
#define DIM        256
#define TWO_N      8192
#define COL_SPLIT  8
#define COLS_PER_WAVE (TWO_N / COL_SPLIT)   /* 1024 */
#define NORM_EPS   1e-8f

// ---------------------------------------------------------------------------
// Kernel 1: L2-normalize each row and store shuffled:
//   dst layout per row: even float2-pairs (k=4t,4t+1) at [2t..2t+1],
//                       odd  float2-pairs (k=4t+2,4t+3) at [128+2t..128+2t+1].
// This makes each lane's WMMA-f32 A/B fragment stream contiguous in memory.
// One block (64 threads) per row; each thread handles 4 consecutive floats.
// ---------------------------------------------------------------------------
__global__ void ntx_normalize_shuffle(const float* __restrict__ z1,
                                      const float* __restrict__ z2,
                                      float* __restrict__ zs) {
  const int row = blockIdx.x;
  const int tid = threadIdx.x;   // 0..63
  const float* src = (row < NSAMP) ? (z1 + (size_t)row * DIM)
                                   : (z2 + (size_t)(row - NSAMP) * DIM);
  const float4 v = ((const float4*)src)[tid];
  float ss = v.x * v.x + v.y * v.y + v.z * v.z + v.w * v.w;

  __shared__ float red[64];
  red[tid] = ss;
  __syncthreads();
  for (int s = 32; s > 0; s >>= 1) {
    if (tid < s) red[tid] += red[tid + s];
    __syncthreads();
  }
  const float norm = fmaxf(sqrtf(red[0]), NORM_EPS);
  const float inv = 1.0f / norm;

  float* dst = zs + (size_t)row * DIM;
  float2 e; e.x = v.x * inv; e.y = v.y * inv;   // even pair (k=4t,4t+1)
  float2 o; o.x = v.z * inv; o.y = v.w * inv;   // odd  pair (k=4t+2,4t+3)
  ((float2*)dst)[tid]         = e;              // float offset 2*tid
  ((float2*)(dst + 128))[tid] = o;              // float offset 128 + 2*tid
}

// ---------------------------------------------------------------------------
// Kernel 2: one wave (32 threads) per block.
//   grid.x = 256 row strips (32 rows each), grid.y = 8 column splits (1024
//   cols each). Wave computes 32x32 sim tiles with 2x2 f32 WMMA accumulators,
//   K=256 in steps of 4 (two steps per global_load_b128), then folds each
//   tile into per-row running sum(exp(2*sim_acc)) and the positive logit.
// A(16x4 f32) and B(4x16 f32) fragments share the same per-lane layout:
//   lanes 0-15  : v0=K+0, v1=K+1 at (row|col) = base + lane
//   lanes 16-31 : v0=K+2, v1=K+3 at (row|col) = base + lane-16
// which the shuffled zs layout serves with contiguous float4 reads.
// ---------------------------------------------------------------------------
__global__ void ntx_sim_lse(const float* __restrict__ zs,
                            float* __restrict__ pSum,
                            float* __restrict__ pPos) {
  const int lane = threadIdx.x;        // 0..31
  const int lm   = lane & 15;
  const int half = lane >> 4;          // 0: K+0/1 stream, 1: K+2/3 stream
  const int rowBase = blockIdx.x * 32;
  const int colBase = blockIdx.y * COLS_PER_WAVE;

  const float* aPtr0 = zs + (size_t)(rowBase + lm)      * DIM + half * 128;
  const float* aPtr1 = zs + (size_t)(rowBase + 16 + lm) * DIM + half * 128;

  float rowSum[2][8];
  float posSum[2][8];
#pragma unroll
  for (int rt = 0; rt < 2; ++rt)
#pragma unroll
    for (int r = 0; r < 8; ++r) { rowSum[rt][r] = 0.0f; posSum[rt][r] = 0.0f; }

  for (int cc = 0; cc < COLS_PER_WAVE; cc += 32) {
    const int c0 = colBase + cc;
    const float* bPtr0 = zs + (size_t)(c0 + lm)      * DIM + half * 128;
    const float* bPtr1 = zs + (size_t)(c0 + 16 + lm) * DIM + half * 128;

    v8f acc00 = {}, acc01 = {}, acc10 = {}, acc11 = {};

#pragma unroll 4
    for (int t = 0; t < 64; t += 2) {       // 64 K-steps of 4; 2 per iter
      const v4f a0 = *(const v4f*)(aPtr0 + 2 * t);
      const v4f a1 = *(const v4f*)(aPtr1 + 2 * t);
      const v4f b0 = *(const v4f*)(bPtr0 + 2 * t);
      const v4f b1 = *(const v4f*)(bPtr1 + 2 * t);
      const v2f a0lo = {a0[0], a0[1]}, a0hi = {a0[2], a0[3]};
      const v2f a1lo = {a1[0], a1[1]}, a1hi = {a1[2], a1[3]};
      const v2f b0lo = {b0[0], b0[1]}, b0hi = {b0[2], b0[3]};
      const v2f b1lo = {b1[0], b1[1]}, b1hi = {b1[2], b1[3]};

      acc00 = __builtin_amdgcn_wmma_f32_16x16x4_f32(false, a0lo, false, b0lo,
                                                    (short)0, acc00, false, false);
      acc01 = __builtin_amdgcn_wmma_f32_16x16x4_f32(false, a0lo, false, b1lo,
                                                    (short)0, acc01, false, false);
      acc10 = __builtin_amdgcn_wmma_f32_16x16x4_f32(false, a1lo, false, b0lo,
                                                    (short)0, acc10, false, false);
      acc11 = __builtin_amdgcn_wmma_f32_16x16x4_f32(false, a1lo, false, b1lo,
                                                    (short)0, acc11, false, false);
      acc00 = __builtin_amdgcn_wmma_f32_16x16x4_f32(false, a0hi, false, b0hi,
                                                    (short)0, acc00, false, false);
      acc01 = __builtin_amdgcn_wmma_f32_16x16x4_f32(false, a0hi, false, b1hi,
                                                    (short)0, acc01, false, false);
      acc10 = __builtin_amdgcn_wmma_f32_16x16x4_f32(false, a1hi, false, b0hi,
                                                    (short)0, acc10, false, false);
      acc11 = __builtin_amdgcn_wmma_f32_16x16x4_f32(false, a1hi, false, b1hi,
                                                    (short)0, acc11, false, false);
    }

    // Fold this 32x32 tile into per-row exp-sums / positives.
    // C/D layout: VGPR r -> M = r + 8*half ; N = lane & 15.
    auto consume = [&](const v8f a, int rt, int ct) {
#pragma unroll
      for (int r = 0; r < 8; ++r) {
        const int row = rowBase + rt * 16 + r + 8 * half;
        const int col = c0 + ct * 16 + lm;
        const float logit = 2.0f * a[r];             // 1/TEMPERATURE = 2
        rowSum[rt][r] += (row != col) ? expf(logit) : 0.0f;
        posSum[rt][r] += (col == (row ^ NSAMP)) ? logit : 0.0f;
      }
    };
    consume(acc00, 0, 0);
    consume(acc01, 0, 1);
    consume(acc10, 1, 0);
    consume(acc11, 1, 1);
  }

  // Reduce across the 16 N-lanes of each half-wave; leaders write partials.
#pragma unroll
  for (int rt = 0; rt < 2; ++rt) {
#pragma unroll
    for (int r = 0; r < 8; ++r) {
      float s = rowSum[rt][r];
      float p = posSum[rt][r];
      for (int m = 1; m < 16; m <<= 1) {
        s += __shfl_xor(s, m, 16);
        p += __shfl_xor(p, m, 16);
      }
      if (lm == 0) {
        const int row = rowBase + rt * 16 + r + 8 * half;
        pSum[(size_t)blockIdx.y * TWO_N + row] = s;
        pPos[(size_t)blockIdx.y * TWO_N + row] = p;
      }
    }
  }
}

// ---------------------------------------------------------------------------
// Kernel 3: deterministic final reduction (single block).
//   out = mean_row( log(sum_splits pSum) - sum_splits pPos )
// ---------------------------------------------------------------------------
__global__ void ntx_finalize(const float* __restrict__ pSum,
                             const float* __restrict__ pPos,
                             float* __restrict__ out) {
  const int tid = threadIdx.x;   // 0..255
  float acc = 0.0f;
  for (int row = tid; row < TWO_N; row += 256) {
    float s = 0.0f, p = 0.0f;
#pragma unroll
    for (int k = 0; k < COL_SPLIT; ++k) {
      s += pSum[(size_t)k * TWO_N + row];
      p += pPos[(size_t)k * TWO_N + row];
    }
    acc += logf(s) - p;
  }
  __shared__ float red[256];
  red[tid] = acc;
  __syncthreads();
  for (int s = 128; s > 0; s >>= 1) {
    if (tid < s) red[tid] += red[tid + s];
    __syncthreads();
  }
  if (tid == 0) out[0] = red[0] * (1.0f / (float)TWO_N);
}

// ---------------------------------------------------------------------------
extern "C" void kernel_launch(void* const* d_in, const int* in_sizes, int n_in,
                              void* d_out, int out_size, void* d_ws, size_t ws_size,
                              hipStream_t stream) {
  const float* z1 = (const float*)d_in[0];
  const float* z2 = (const float*)d_in[1];

  float* zs   = (float*)d_ws;                         // 8192*256 = 8 MB
  float* pSum = zs + (size_t)TWO_N * DIM;             // 8*8192 floats
  float* pPos = pSum + (size_t)COL_SPLIT * TWO_N;     // 8*8192 floats

  ntx_normalize_shuffle<<<TWO_N, 64, 0, stream>>>(z1, z2, zs);

  dim3 grid(TWO_N / 32, COL_SPLIT);                   // 256 x 8 waves
  ntx_sim_lse<<<grid, 32, 0, stream>>>(zs, pSum, pPos);

  ntx_finalize<<<1, 256, 0, stream>>>(pSum, pPos, (float*)d_out);
}